// GCNBaseline_38027640439269
// MI455X (gfx1250) — compile-verified
//
#include <hip/hip_runtime.h>
#include <hip/hip_bf16.h>

// -------------------------------------------------------------------------
// 2-layer GCN (PyG GCNConv) for MI455X / gfx1250.
//
// Strategy (reasoned from MI455X specs, compile-only):
//  * Aggregation dominates (~600MB gather vs 1.5 GFLOP dense). The naive
//    scatter needs 128M f32 atomics -> atomic-unit bound. Instead build
//    CSR-by-dst per call (2 edge passes, int atomics only) and aggregate
//    as a pure gather: zero float atomics, bandwidth-bound (~tens of us
//    at 23.3 TB/s).
//  * Dense transforms: V_WMMA_F32_16X16X4_F32 (full fp32 precision).
//    W (<=36KB) is staged once per workgroup into LDS via the CDNA5 async
//    global->LDS path (global_load_async_to_lds_b128 + s_wait_asynccnt);
//    each wave computes a 16-row x N strip, reusing the A fragment across
//    all N-tiles and serving B fragments from LDS.
// -------------------------------------------------------------------------

#define N_NODES 50000
#define N_EDGES 800000
#define C_IN    96
#define C_HID   96
#define C_OUT   64

typedef __attribute__((ext_vector_type(2))) float v2f;
typedef __attribute__((ext_vector_type(8))) float v8f;

// ---------------- CSR build ----------------

__global__ void k_zero_int(int* __restrict__ p, int n) {
  int i = blockIdx.x * blockDim.x + threadIdx.x;
  if (i < n) p[i] = 0;
}

__global__ void k_count(const int* __restrict__ dst, int* __restrict__ count, int e) {
  int i = blockIdx.x * blockDim.x + threadIdx.x;
  if (i < e) atomicAdd(&count[dst[i]], 1);
}

// dinv[i] = rsqrt(in_degree + 1 self-loop)  (always > 0)
__global__ void k_dinv(const int* __restrict__ count, float* __restrict__ dinv, int n) {
  int i = blockIdx.x * blockDim.x + threadIdx.x;
  if (i < n) dinv[i] = __frsqrt_rn((float)(count[i] + 1));
}

// single-workgroup exclusive scan over `count` -> rowstart[0..n], cursor copy
__global__ void k_scan(const int* __restrict__ count, int* __restrict__ rowstart,
                       int* __restrict__ cursor, int n) {
  __shared__ int part[1024];
  const int tid = threadIdx.x;
  const int chunk = (n + 1023) / 1024;
  const int base = tid * chunk;
  int s = 0;
  for (int j = 0; j < chunk; ++j) {
    int idx = base + j;
    if (idx < n) s += count[idx];
  }
  part[tid] = s;
  __syncthreads();
  if (tid == 0) {
    int acc = 0;
    for (int i = 0; i < 1024; ++i) { int v = part[i]; part[i] = acc; acc += v; }
    rowstart[n] = acc;
  }
  __syncthreads();
  int acc = part[tid];
  for (int j = 0; j < chunk; ++j) {
    int idx = base + j;
    if (idx < n) { rowstart[idx] = acc; cursor[idx] = acc; acc += count[idx]; }
  }
}

__global__ void k_fill(const int* __restrict__ src, const int* __restrict__ dst,
                       int* __restrict__ cursor, int* __restrict__ srcs_sorted, int e) {
  int i = blockIdx.x * blockDim.x + threadIdx.x;
  if (i < e) {
    int p = atomicAdd(&cursor[dst[i]], 1);
    srcs_sorted[p] = src[i];
  }
}

// ---------------- dense GEMM via f32 WMMA, W staged in LDS ----------------
// D(M x NT*16) = A(M x K) * Bw(K x NT*16).  M,K multiples of 16, K mult of 4.
// Block = 256 threads (8 waves); each wave owns one 16-row strip, all N tiles.

template <int NT>
__global__ __launch_bounds__(256) void k_gemm_wmma(const float* __restrict__ A,
                                                   const float* __restrict__ Bw,
                                                   float* __restrict__ D,
                                                   int Mtiles, int K) {
  constexpr int N = NT * 16;
  __shared__ float sW[96 * N];  // K <= 96

  // ---- async stage W into LDS (16B per lane per issue) ----
  const int tid = threadIdx.x;
  const int total = K * N;  // multiple of blockDim*4 for our shapes
  for (int i = tid * 4; i < total; i += blockDim.x * 4) {
    const float* g = Bw + i;
    unsigned lds = (unsigned)(size_t)(&sW[i]);  // flat LDS addr low 32 bits
    asm volatile("global_load_async_to_lds_b128 %0, %1, off"
                 :: "v"(lds), "v"(g) : "memory");
  }
  asm volatile("s_wait_asynccnt 0" ::: "memory");
  __syncthreads();

  const int lane = tid & 31;
  const int wave = tid >> 5;
  const int tm   = blockIdx.x * 8 + wave;   // wave-uniform
  if (tm >= Mtiles) return;                 // EXEC stays all-1s below

  const int half = lane >> 4;  // 0/1
  const int lr   = lane & 15;
  const int m0   = tm << 4;

  v8f acc[NT];
#pragma unroll
  for (int t = 0; t < NT; ++t) acc[t] = (v8f){};

  for (int k0 = 0; k0 < K; k0 += 4) {
    // A fragment: lanes 0-15 -> K=k0,k0+1 ; lanes 16-31 -> K=k0+2,k0+3
    const float* ap = A + (size_t)(m0 + lr) * K + (k0 + 2 * half);
    v2f av;
    av.x = ap[0];
    av.y = ap[1];
    const int kb = (k0 + 2 * half) * N + lr;
#pragma unroll
    for (int t = 0; t < NT; ++t) {
      v2f bv;
      bv.x = sW[kb + 16 * t];
      bv.y = sW[kb + N + 16 * t];
      acc[t] = __builtin_amdgcn_wmma_f32_16x16x4_f32(false, av, false, bv,
                                                     (short)0, acc[t], false, false);
    }
  }

#pragma unroll
  for (int t = 0; t < NT; ++t) {
#pragma unroll
    for (int r = 0; r < 8; ++r) {
      D[(size_t)(m0 + r + 8 * half) * N + 16 * t + lr] = acc[t][r];
    }
  }
}

// ---------------- gather-based aggregation (no float atomics) ----------------
// out[i,:] = bias + dinv[i] * ( dinv[i]*h[i,:] + sum_{s in inEdges(i)} dinv[s]*h[s,:] )
// optionally fused ReLU.  One thread per (node, 4 channels).

__global__ void k_aggregate(const float* __restrict__ h,
                            const float* __restrict__ dinv,
                            const int* __restrict__ rowstart,
                            const int* __restrict__ srcs,
                            const float* __restrict__ bias,
                            float* __restrict__ out,
                            int C, int n, int do_relu) {
  const int C4 = C >> 2;
  long long t = (long long)blockIdx.x * blockDim.x + threadIdx.x;
  if (t >= (long long)n * C4) return;
  const int c4 = (int)(t % C4);
  const int i  = (int)(t / C4);

  const float di = dinv[i];
  const float4 hv = *(const float4*)(h + (size_t)i * C + 4 * c4);
  float sx = di * hv.x, sy = di * hv.y, sz = di * hv.z, sw = di * hv.w;

  const int b = rowstart[i];
  const int e = rowstart[i + 1];
  for (int j = b; j < e; ++j) {
    const int s = srcs[j];
    const float w = dinv[s];
    const float4 v = *(const float4*)(h + (size_t)s * C + 4 * c4);
    sx += w * v.x; sy += w * v.y; sz += w * v.z; sw += w * v.w;
  }

  const float4 bb = *(const float4*)(bias + 4 * c4);
  float ox = di * sx + bb.x;
  float oy = di * sy + bb.y;
  float oz = di * sz + bb.z;
  float ow = di * sw + bb.w;
  if (do_relu) {
    ox = fmaxf(ox, 0.0f); oy = fmaxf(oy, 0.0f);
    oz = fmaxf(oz, 0.0f); ow = fmaxf(ow, 0.0f);
  }
  float4* o = (float4*)(out + (size_t)i * C + 4 * c4);
  *o = make_float4(ox, oy, oz, ow);
}

// ---------------- launch ----------------

static inline int cdiv(long long a, int b) { return (int)((a + b - 1) / b); }

extern "C" void kernel_launch(void* const* d_in, const int* in_sizes, int n_in,
                              void* d_out, int out_size, void* d_ws, size_t ws_size,
                              hipStream_t stream) {
  const float* x  = (const float*)d_in[0];
  const int*   ei = (const int*)d_in[1];   // [2, E] int32
  const float* W1 = (const float*)d_in[2];
  const float* b1 = (const float*)d_in[3];
  const float* W2 = (const float*)d_in[4];
  const float* b2 = (const float*)d_in[5];
  float*       out = (float*)d_out;

  const int* src = ei;
  const int* dst = ei + N_EDGES;

  // workspace layout (16B-aligned regions)
  char* w = (char*)d_ws;
  int*   count    = (int*)w;    w += (size_t)N_NODES * 4;        // 200000
  int*   rowstart = (int*)w;    w += (size_t)(N_NODES + 4) * 4;  // 200016
  int*   cursor   = (int*)w;    w += (size_t)N_NODES * 4;        // 200000
  int*   srcs_s   = (int*)w;    w += (size_t)N_EDGES * 4;        // 3200000
  float* dinv     = (float*)w;  w += (size_t)N_NODES * 4;        // 200000
  float* h        = (float*)w;  w += (size_t)N_NODES * C_HID * 4;
  float* a        = (float*)w;

  const int B = 256;

  // ---- CSR build + normalization (rebuilt every call; ws is not preserved) ----
  k_zero_int<<<cdiv(N_NODES, B), B, 0, stream>>>(count, N_NODES);
  k_count<<<cdiv(N_EDGES, B), B, 0, stream>>>(dst, count, N_EDGES);
  k_dinv<<<cdiv(N_NODES, B), B, 0, stream>>>(count, dinv, N_NODES);
  k_scan<<<1, 1024, 0, stream>>>(count, rowstart, cursor, N_NODES);
  k_fill<<<cdiv(N_EDGES, B), B, 0, stream>>>(src, dst, cursor, srcs_s, N_EDGES);

  // ---- layer 1: h = X*W1 ; a = relu(aggregate(h) + b1) ----
  k_gemm_wmma<6><<<cdiv(N_NODES / 16, 8), 256, 0, stream>>>(x, W1, h,
                                                            N_NODES / 16, C_IN);
  k_aggregate<<<cdiv((long long)N_NODES * (C_HID / 4), B), B, 0, stream>>>(
      h, dinv, rowstart, srcs_s, b1, a, C_HID, N_NODES, 1);

  // ---- layer 2: h2 = a*W2 ; out = aggregate(h2) + b2 ----
  k_gemm_wmma<4><<<cdiv(N_NODES / 16, 8), 256, 0, stream>>>(a, W2, h,
                                                            N_NODES / 16, C_HID);
  k_aggregate<<<cdiv((long long)N_NODES * (C_OUT / 4), B), B, 0, stream>>>(
      h, dinv, rowstart, srcs_s, b2, out, C_OUT, N_NODES, 0);
}